// BatchTCLoss_64338610094317
// MI455X (gfx1250) — compile-verified
//
#include <hip/hip_runtime.h>
#include <math.h>

// Problem constants (from reference)
#define BB   512                 // batch
#define ZZ   256                 // latent dim
#define NPIX (512 * 3 * 64 * 64) // 6,291,456 pixels
#define LOG2PI 1.8378770664093453f

// Workspace layout (float element offsets)
#define WS_ALPHA 0
#define WS_BETA  (BB * ZZ)
#define WS_GAMMA (2 * BB * ZZ)
#define WS_G     (3 * BB * ZZ)
#define WS_DWKL  (WS_G + BB)
#define WS_BCE   (WS_DWKL + BB)
#define WS_S     (WS_BCE + 512)
#define WS_M     (WS_S + BB * BB)
#define WS_LOGQZ (WS_M + BB * ZZ)
// total = WS_LOGQZ + BB = 788,480 floats ~= 3.16 MB

typedef float v2f __attribute__((ext_vector_type(2)));
typedef float v8f __attribute__((ext_vector_type(8)));

__device__ __forceinline__ float block_reduce_sum_256(float v, float* sm) {
    int t = threadIdx.x;
    sm[t] = v;
    __syncthreads();
    for (int s = 128; s > 0; s >>= 1) {
        if (t < s) sm[t] += sm[t + s];
        __syncthreads();
    }
    float r = sm[0];
    __syncthreads();
    return r;
}

__device__ __forceinline__ float block_reduce_max_256(float v, float* sm) {
    int t = threadIdx.x;
    sm[t] = v;
    __syncthreads();
    for (int s = 128; s > 0; s >>= 1) {
        if (t < s) sm[t] = fmaxf(sm[t], sm[t + s]);
        __syncthreads();
    }
    float r = sm[0];
    __syncthreads();
    return r;
}

// ---------------------------------------------------------------------------
// Kernel 1: per-(j,k) coefficients + per-j gamma sum G[j] + dw-KL partials.
// grid = BB blocks (one per j), block = ZZ threads (one per k).
// ---------------------------------------------------------------------------
__global__ void prep_kernel(const float* __restrict__ mu,
                            const float* __restrict__ logvar,
                            float* __restrict__ alpha,
                            float* __restrict__ beta,
                            float* __restrict__ gamma,
                            float* __restrict__ G,
                            float* __restrict__ dwkl_part) {
    __shared__ float sm[256];
    int j = blockIdx.x;
    int k = threadIdx.x;
    int idx = j * ZZ + k;
    float m  = mu[idx];
    float lv = logvar[idx];
    float w  = expf(lv);             // NOTE: faithful to source quirk exp(+logvar)
    float a  = -0.5f * w;
    float b  = m * w;
    float g  = -0.5f * (m * m * w + lv + LOG2PI);
    alpha[idx] = a;
    beta[idx]  = b;
    gamma[idx] = g;
    float dw = -0.5f * lv + 0.5f * expf(lv + m * m) - 0.5f;

    float gsum = block_reduce_sum_256(g, sm);
    float dsum = block_reduce_sum_256(dw, sm);
    if (k == 0) {
        G[j] = gsum;
        dwkl_part[j] = dsum;
    }
}

// ---------------------------------------------------------------------------
// Kernel 2: BCE partial sums. grid = 512 blocks x 256 threads, grid-stride.
// Deterministic: fixed grid, in-block tree reduce, one partial per block.
// ---------------------------------------------------------------------------
__global__ void bce_kernel(const float* __restrict__ data,
                           const float* __restrict__ recon,
                           float* __restrict__ bce_part) {
    __shared__ float sm[256];
    float acc = 0.0f;
    int stride = gridDim.x * blockDim.x; // 131072 -> 48 iters
    for (int i = blockIdx.x * blockDim.x + threadIdx.x; i < NPIX; i += stride) {
        float d = data[i];
        float r = recon[i];
        acc += d * logf(r) + (1.0f - d) * logf(1.0f - r);
    }
    float s = block_reduce_sum_256(acc, sm);
    if (threadIdx.x == 0) bce_part[blockIdx.x] = s;
}

// ---------------------------------------------------------------------------
// Kernel 3: S[i,j] = sum_k logq(i,j,k) via two f32 WMMA GEMM chains.
//   S = (s^2) * alpha^T + s * beta^T + G[j]
// One wave (32 threads) per 16x16 output tile; 64 k-steps of K=4.
// VGPR layouts per ISA 7.12.2:
//   A (16x4 f32):  lane L -> row m = L%16, holds K = (L/16)*2 + v, v=0,1
//   B (4x16 f32):  lane L -> col n = L%16, holds K = (L/16)*2 + v, v=0,1
//   C/D (16x16):   VGPR r -> row = r + (L/16)*8, col = L%16
// ---------------------------------------------------------------------------
__global__ __launch_bounds__(32)
void logq_gemm_kernel(const float* __restrict__ sample,
                      const float* __restrict__ alpha,
                      const float* __restrict__ beta,
                      const float* __restrict__ G,
                      float* __restrict__ S) {
    const int lane = threadIdx.x;        // 0..31
    const int n    = lane & 15;          // row (A) / col (B)
    const int kh   = (lane >> 4) << 1;   // 0 or 2
    const int j0   = blockIdx.x * 16;
    const int i0   = blockIdx.y * 16;

    const float* arow = sample + (i0 + n) * ZZ + kh; // A row for this lane
    const float* brA  = alpha  + (j0 + n) * ZZ + kh; // B (alpha^T) col
    const float* brB  = beta   + (j0 + n) * ZZ + kh; // B (beta^T)  col

    v8f acc1 = {};
    v8f acc2 = {};
    for (int k0 = 0; k0 < ZZ; k0 += 4) {
        float2 av  = *(const float2*)(arow + k0);
        float2 bav = *(const float2*)(brA + k0);
        float2 bbv = *(const float2*)(brB + k0);
        v2f a_s  = {av.x, av.y};
        v2f a_s2 = {av.x * av.x, av.y * av.y};
        v2f b_a  = {bav.x, bav.y};
        v2f b_b  = {bbv.x, bbv.y};
        // D = A*B + C  (8-arg form: neg_a, A, neg_b, B, c_mod, C, reuse_a, reuse_b)
        acc1 = __builtin_amdgcn_wmma_f32_16x16x4_f32(false, a_s2, false, b_a,
                                                     (short)0, acc1, false, false);
        acc2 = __builtin_amdgcn_wmma_f32_16x16x4_f32(false, a_s,  false, b_b,
                                                     (short)0, acc2, false, false);
    }

    float gj = G[j0 + n];
    int half = lane >> 4;
    for (int r = 0; r < 8; ++r) {
        int row = i0 + r + half * 8;
        S[row * BB + j0 + n] = acc1[r] + acc2[r] + gj;
    }
}

// ---------------------------------------------------------------------------
// Kernel 4: logqz[i] = logsumexp_j S[i,j]. One block per row.
// ---------------------------------------------------------------------------
__global__ void lse_rows_kernel(const float* __restrict__ S,
                                float* __restrict__ logqz) {
    __shared__ float sm[256];
    int i = blockIdx.x;
    int t = threadIdx.x;
    const float* row = S + i * BB;
    float v0 = row[t];
    float v1 = row[t + 256];
    float m = block_reduce_max_256(fmaxf(v0, v1), sm);
    float e = expf(v0 - m) + expf(v1 - m);
    float s = block_reduce_sum_256(e, sm);
    if (t == 0) logqz[i] = m + logf(s);
}

// ---------------------------------------------------------------------------
// Kernel 5: M[i,k] = logsumexp_j (alpha[j,k]*s^2 + beta[j,k]*s + gamma[j,k]).
// Online LSE over j. Block = (64 k, 4 i); coefficients stream through L2
// (1.5 MB working set, fully L2-resident on 192 MB).
// ---------------------------------------------------------------------------
__global__ void lse_jk_kernel(const float* __restrict__ sample,
                              const float* __restrict__ alpha,
                              const float* __restrict__ beta,
                              const float* __restrict__ gamma,
                              float* __restrict__ M) {
    int k = blockIdx.x * 64 + threadIdx.x;
    int i = blockIdx.y * 4 + threadIdx.y;
    float s  = sample[i * ZZ + k];
    float s2 = s * s;
    float m = -INFINITY;
    float acc = 0.0f;
    for (int j = 0; j < BB; ++j) {
        int idx = j * ZZ + k;
        float lq = fmaf(alpha[idx], s2, fmaf(beta[idx], s, gamma[idx]));
        float nm = fmaxf(m, lq);
        acc = acc * expf(m - nm) + expf(lq - nm);
        m = nm;
    }
    M[i * ZZ + k] = m + logf(acc);
}

// ---------------------------------------------------------------------------
// Kernel 6: final combine. Single block, deterministic ordered reductions.
// ---------------------------------------------------------------------------
__global__ void final_kernel(const float* __restrict__ bce_part,
                             const float* __restrict__ dwkl_part,
                             const float* __restrict__ logqz,
                             const float* __restrict__ M,
                             const int* __restrict__ dataset_size,
                             float* __restrict__ out) {
    __shared__ float sm[256];
    int t = threadIdx.x;

    float log_norm = logf((float)BB) + logf((float)(*dataset_size));

    float bce_sum = block_reduce_sum_256(bce_part[t] + bce_part[t + 256], sm);
    float dw_sum  = block_reduce_sum_256(dwkl_part[t] + dwkl_part[t + 256], sm);

    float tcacc = 0.0f;
    for (int i = t; i < BB; i += 256) {
        const float* Mrow = M + i * ZZ;
        float rs = 0.0f;
        for (int k = 0; k < ZZ; ++k) rs += Mrow[k];
        float lqz  = logqz[i] - log_norm;            // logqz
        float lqpm = rs - (float)ZZ * log_norm;      // logqz_prodmarginals
        tcacc += lqz - lqpm;
    }
    float tc_sum = block_reduce_sum_256(tcacc, sm);

    if (t == 0) {
        float rec_loss = -bce_sum / (float)BB;
        float dw_kl    = dw_sum / (float)BB;
        float tc_loss  = tc_sum / (float)BB;
        out[0] = rec_loss + tc_loss + dw_kl;
    }
}

// ---------------------------------------------------------------------------
extern "C" void kernel_launch(void* const* d_in, const int* in_sizes, int n_in,
                              void* d_out, int out_size, void* d_ws, size_t ws_size,
                              hipStream_t stream) {
    const float* data    = (const float*)d_in[0];
    const float* recon   = (const float*)d_in[1];
    const float* sample  = (const float*)d_in[2];
    const float* mu      = (const float*)d_in[3];
    const float* logvar  = (const float*)d_in[4];
    const int*   dsz     = (const int*)d_in[5];
    float* ws  = (float*)d_ws;
    float* out = (float*)d_out;

    prep_kernel<<<BB, 256, 0, stream>>>(mu, logvar,
                                        ws + WS_ALPHA, ws + WS_BETA, ws + WS_GAMMA,
                                        ws + WS_G, ws + WS_DWKL);

    bce_kernel<<<512, 256, 0, stream>>>(data, recon, ws + WS_BCE);

    logq_gemm_kernel<<<dim3(BB / 16, BB / 16), 32, 0, stream>>>(
        sample, ws + WS_ALPHA, ws + WS_BETA, ws + WS_G, ws + WS_S);

    lse_rows_kernel<<<BB, 256, 0, stream>>>(ws + WS_S, ws + WS_LOGQZ);

    lse_jk_kernel<<<dim3(ZZ / 64, BB / 4), dim3(64, 4), 0, stream>>>(
        sample, ws + WS_ALPHA, ws + WS_BETA, ws + WS_GAMMA, ws + WS_M);

    final_kernel<<<1, 256, 0, stream>>>(ws + WS_BCE, ws + WS_DWKL,
                                        ws + WS_LOGQZ, ws + WS_M, dsz, out);
}